// KF_20332375179766
// MI455X (gfx1250) — compile-verified
//
#include <hip/hip_runtime.h>
#include <math.h>

// Problem constants (match reference)
static constexpr int Bn = 2048;
static constexpr int Tn = 1000;

typedef float v2f __attribute__((ext_vector_type(2)));
typedef float v4f __attribute__((ext_vector_type(4)));
typedef float v8f __attribute__((ext_vector_type(8)));

// ---------------------------------------------------------------------------
// Kernel 1: serial Riccati recursion (single 64-thread block, thread = (i,j)).
// Emits, per step t:
//   - Ps[t]  (8x8 posterior covariance) straight into d_out tail
//   - Gws[t] : 256 floats = the WMMA *A-operand* for the 16x16 matrix
//       G_t rows 0..7  = [ (I-K H)F | K ]          (real recurrence)
//       G_t rows 8..13 = copies of rows 2..7        (re-materializes states
//                        2..7 into lanes 16..31 of the WMMA D result, so the
//                        scan kernel needs NO cross-lane shuffles at all)
//       G_t rows 14,15 = 0
//     pre-swizzled for V_WMMA_F32_16X16X4_F32 A-operand:
//       lane L, slice s(0..3), pair p(0..1):
//          m = L&15 ; k = 4s + (L<16 ? p : 2+p) ; val = G_t[m][k]
// Stops early once P converges bitwise in f32; records t_stop.
// ---------------------------------------------------------------------------
__global__ __launch_bounds__(64) void kf_gain_kernel(
    const float* __restrict__ F, const float* __restrict__ Gmat,
    const float* __restrict__ H, const float* __restrict__ Q,
    const float* __restrict__ R, const float* __restrict__ P0,
    float* __restrict__ Gws, int* __restrict__ t_stop_ptr,
    float* __restrict__ Ps)
{
    __shared__ float Fm[8][8], Hm[8][8], Qe[8][8], Rm[8][8];
    __shared__ float P[8][8], Pneg[8][8], T1[8][8], T2[8][8];
    __shared__ float S[8][8], X[8][8], IKH[8][8], Amat[8][8], Pn[8][8];
    __shared__ float GQ[8][8];
    __shared__ int flag;

    const int tid = threadIdx.x;
    const int i = tid >> 3, j = tid & 7;

    Fm[i][j] = F[tid];
    Hm[i][j] = H[tid];
    Rm[i][j] = R[tid];
    P[i][j]  = P0[tid];
    __syncthreads();

    // Qe = G Q G^T
    float acc = 0.f;
    #pragma unroll
    for (int k = 0; k < 8; ++k) acc += Gmat[i * 8 + k] * Q[k * 8 + j];
    GQ[i][j] = acc;
    __syncthreads();
    acc = 0.f;
    #pragma unroll
    for (int k = 0; k < 8; ++k) acc += GQ[i][k] * Gmat[j * 8 + k];
    Qe[i][j] = acc;
    __syncthreads();

    int tstop = Tn;
    for (int t = 0; t < Tn; ++t) {
        // T1 = F * P
        acc = 0.f;
        #pragma unroll
        for (int k = 0; k < 8; ++k) acc += Fm[i][k] * P[k][j];
        T1[i][j] = acc;
        __syncthreads();
        // Pneg = T1 * F^T + Qe
        acc = Qe[i][j];
        #pragma unroll
        for (int k = 0; k < 8; ++k) acc += T1[i][k] * Fm[j][k];
        Pneg[i][j] = acc;
        __syncthreads();
        // T2 = H * Pneg
        acc = 0.f;
        #pragma unroll
        for (int k = 0; k < 8; ++k) acc += Hm[i][k] * Pneg[k][j];
        T2[i][j] = acc;
        __syncthreads();
        // S = T2 * H^T + R
        acc = Rm[i][j];
        #pragma unroll
        for (int k = 0; k < 8; ++k) acc += T2[i][k] * Hm[j][k];
        S[i][j] = acc;
        __syncthreads();
        // Cholesky (lower, in place in S)
        #pragma unroll 1
        for (int k = 0; k < 8; ++k) {
            if (i == k && j == k) S[k][k] = sqrtf(S[k][k]);
            __syncthreads();
            if (j == k && i > k) S[i][k] = S[i][k] / S[k][k];
            __syncthreads();
            if (i > k && j > k && j <= i) S[i][j] -= S[i][k] * S[j][k];
            __syncthreads();
        }
        // Forward solve L Z = T2  (Z -> X)
        #pragma unroll 1
        for (int r = 0; r < 8; ++r) {
            if (i == r) {
                float s = T2[r][j];
                for (int m = 0; m < r; ++m) s -= S[r][m] * X[m][j];
                X[r][j] = s / S[r][r];
            }
            __syncthreads();
        }
        // Back solve L^T W = Z  (W -> X).  X = S^{-1} H Pneg, K = X^T.
        #pragma unroll 1
        for (int r = 7; r >= 0; --r) {
            if (i == r) {
                float s = X[r][j];
                for (int m = r + 1; m < 8; ++m) s -= S[m][r] * X[m][j];
                X[r][j] = s / S[r][r];
            }
            __syncthreads();
        }
        // IKH = I - K H   (K[i][m] = X[m][i])
        acc = (i == j) ? 1.f : 0.f;
        #pragma unroll
        for (int m = 0; m < 8; ++m) acc -= X[m][i] * Hm[m][j];
        IKH[i][j] = acc;
        __syncthreads();
        // Amat = IKH * F ; Pn = IKH * Pneg
        acc = 0.f;
        float pacc = 0.f;
        #pragma unroll
        for (int k = 0; k < 8; ++k) {
            acc  += IKH[i][k] * Fm[k][j];
            pacc += IKH[i][k] * Pneg[k][j];
        }
        Amat[i][j] = acc;
        Pn[i][j]   = pacc;
        __syncthreads();
        // Symmetrize P
        float sym = 0.5f * (Pn[i][j] + Pn[j][i]);
        __syncthreads();
        Pn[i][j] = sym;
        __syncthreads();

        if (tid == 0) flag = 0;
        __syncthreads();
        if (Pn[i][j] != P[i][j]) flag = 1;

        // Outputs
        Ps[t * 64 + tid] = Pn[i][j];
        float* gout = Gws + (size_t)t * 256;
        #pragma unroll
        for (int q = 0; q < 4; ++q) {
            int f    = tid * 4 + q;
            int lane = f >> 3, slot = f & 7;
            int s2 = slot >> 1, p = slot & 1;
            int m = lane & 15;
            int k = 4 * s2 + ((lane < 16) ? p : (2 + p));
            // Row duplication map: rows 8..13 replicate rows 2..7
            int mm = (m < 8) ? m : ((m <= 13) ? (m - 6) : -1);
            float v = 0.f;
            if (mm >= 0) v = (k < 8) ? Amat[mm][k] : X[k - 8][mm];
            gout[f] = v;
        }
        P[i][j] = Pn[i][j];
        __syncthreads();
        if (flag == 0) { tstop = t + 1; break; }
    }
    if (tid == 0) *t_stop_ptr = tstop;
}

// ---------------------------------------------------------------------------
// Kernel 2: replicate converged gains/covariances for t >= t_stop.
// ---------------------------------------------------------------------------
__global__ __launch_bounds__(256) void kf_fill_kernel(
    float* __restrict__ Gws, const int* __restrict__ t_stop_ptr,
    float* __restrict__ Ps)
{
    const int t  = blockIdx.x;
    const int q  = threadIdx.x;
    const int ts = *t_stop_ptr;
    if (t >= ts && ts >= 1) {
        const int src = ts - 1;
        Gws[(size_t)t * 256 + q] = Gws[(size_t)src * 256 + q];
        if (q < 64) Ps[t * 64 + q] = Ps[src * 64 + q];
    }
}

// ---------------------------------------------------------------------------
// Kernel 3: batched state recursion via V_WMMA_F32_16X16X4_F32 (wave32).
// One wave per 16 batch rows (N dim).  x^T carried in D/C layout:
//   lane n (0..15)  VGPR r = x[n][r]          (rows 0..7 of D)
//   lane n+16       VGPR r = x[n][r+2]  r<6   (rows 8..13 = duplicated rows)
// so both the B-operand feedback and the trajectory store are per-lane:
// NO ds_bpermute / shuffles anywhere.  Per step: y-half WMMAs first (no
// dependence on previous state), then the two x-half WMMAs into separate
// accumulators -> critical path = 1 WMMA + one vector add.
// ---------------------------------------------------------------------------
__global__ __launch_bounds__(32) void kf_scan_kernel(
    const float* __restrict__ Y, const float* __restrict__ x0,
    const float* __restrict__ Gws, float* __restrict__ traj)
{
    const int lane = threadIdx.x;
    const int n    = lane & 15;
    const bool lo  = lane < 16;
    const size_t row = (size_t)(blockIdx.x * 16 + n);

    const float* yrow = Y    + row * (size_t)Tn * 8 + (lo ? 0 : 2);
    float*       orow = traj + row * (size_t)Tn * 8 + (lo ? 0 : 4);

    // init x^T in the duplicated-row register convention
    v4f xa = *(const v4f*)(x0 + row * 8);
    v4f xb = *(const v4f*)(x0 + row * 8 + 4);
    v8f xv;
    xv[0] = lo ? xa.x : xa.z;   // lo: x0   hi: x2
    xv[1] = lo ? xa.y : xa.w;   // lo: x1   hi: x3
    xv[2] = lo ? xa.z : xb.x;   // lo: x2   hi: x4
    xv[3] = lo ? xa.w : xb.y;   // lo: x3   hi: x5
    xv[4] = lo ? xb.x : xb.z;   // lo: x4   hi: x6
    xv[5] = lo ? xb.y : xb.w;   // lo: x5   hi: x7
    xv[6] = lo ? xb.z : 0.f;
    xv[7] = lo ? xb.w : 0.f;

    const v8f zero8 = {0.f, 0.f, 0.f, 0.f, 0.f, 0.f, 0.f, 0.f};

    for (int t = 0; t < Tn; ++t) {
        // Gain A-operands: 32B per lane, pre-swizzled by kernel 1.
        const v4f* gp = (const v4f*)(Gws + (size_t)t * 256 + lane * 8);
        v4f g0 = gp[0];
        v4f g1 = gp[1];
        v2f a0 = {g0.x, g0.y};   // K slice 0..3   (x half)
        v2f a1 = {g0.z, g0.w};   // K slice 4..7   (x half)
        v2f a2 = {g1.x, g1.y};   // K slice 8..11  (y half)
        v2f a3 = {g1.z, g1.w};   // K slice 12..15 (y half)

        // y^T B-operand: lane<16 holds obs {0,1},{4,5}; lane>=16 {2,3},{6,7}
        v2f b2 = *(const v2f*)(yrow + (size_t)t * 8);
        v2f b3 = *(const v2f*)(yrow + (size_t)t * 8 + 4);

        // x^T B-operand: purely per-lane thanks to duplicated rows
        v2f b0 = {xv[0], xv[1]};   // lo: K rows 0,1   hi: K rows 2,3
        v2f b1 = {xv[4], xv[5]};   // lo: K rows 4,5   hi: K rows 6,7

        // y-half first (independent of previous state), then the two x-half
        // WMMAs into separate accumulators to halve the dependent chain.
        v8f cy = __builtin_amdgcn_wmma_f32_16x16x4_f32(false, a2, false, b2,
                                                       (short)0, zero8, false, false);
        cy = __builtin_amdgcn_wmma_f32_16x16x4_f32(false, a3, false, b3,
                                                   (short)0, cy, false, false);
        v8f c1 = __builtin_amdgcn_wmma_f32_16x16x4_f32(false, a0, false, b0,
                                                       (short)0, cy, false, false);
        v8f c2 = __builtin_amdgcn_wmma_f32_16x16x4_f32(false, a1, false, b1,
                                                       (short)0, zero8, false, false);
        xv = c1 + c2;

        // Store x_t: lane n -> states 0..3 ; lane n+16 -> states 4..7
        // (hi lanes: xv[2..5] = states 4..7 via duplicated rows)
        v4f out;
        out.x = lo ? xv[0] : xv[2];
        out.y = lo ? xv[1] : xv[3];
        out.z = lo ? xv[2] : xv[4];
        out.w = lo ? xv[3] : xv[5];
        *(v4f*)(orow + (size_t)t * 8) = out;
    }
}

// ---------------------------------------------------------------------------
extern "C" void kernel_launch(void* const* d_in, const int* in_sizes, int n_in,
                              void* d_out, int out_size, void* d_ws, size_t ws_size,
                              hipStream_t stream) {
    const float* Y    = (const float*)d_in[0];
    const float* F    = (const float*)d_in[1];
    const float* Gmat = (const float*)d_in[2];
    const float* H    = (const float*)d_in[3];
    const float* Q    = (const float*)d_in[4];
    const float* R    = (const float*)d_in[5];
    const float* x0   = (const float*)d_in[6];
    const float* P0   = (const float*)d_in[7];

    float* traj = (float*)d_out;
    float* Ps   = traj + (size_t)Bn * Tn * 8;

    int*   t_stop = (int*)d_ws;
    float* Gws    = (float*)((char*)d_ws + 256);

    kf_gain_kernel<<<1, 64, 0, stream>>>(F, Gmat, H, Q, R, P0, Gws, t_stop, Ps);
    kf_fill_kernel<<<Tn, 256, 0, stream>>>(Gws, t_stop, Ps);
    kf_scan_kernel<<<Bn / 16, 32, 0, stream>>>(Y, x0, Gws, traj);
}